// CubemapEncoder_14053132993130
// MI455X (gfx1250) — compile-verified
//
#include <hip/hip_runtime.h>

typedef float v2f __attribute__((ext_vector_type(2)));
typedef float v4f __attribute__((ext_vector_type(4)));
typedef int   i32x4 __attribute__((vector_size(16)));   // matches builtin's param type

#define AS1 __attribute__((address_space(1)))
#define AS3 __attribute__((address_space(3)))

#define LRES   256
#define CCH    6
#define NFACE  6
#define PADC   8           // repacked texel = 8 floats (6 channels + 2 pad) = 32 B aligned
#define NPIX   (NFACE * LRES * LRES)

// gfx1250 async Global<->LDS copy path (ASYNCcnt), if this toolchain exposes it
#define GFX1250_HAS_ASYNC ( \
    __has_builtin(__builtin_amdgcn_global_load_async_to_lds_b32) && \
    __has_builtin(__builtin_amdgcn_global_store_async_from_lds_b128) && \
    __has_builtin(__builtin_amdgcn_s_wait_asynccnt))

__device__ __forceinline__ int clampi(int v, int lo, int hi) {
    return v < lo ? lo : (v > hi ? hi : v);
}

// ---------------------------------------------------------------------------
// Pass 1: repack (6, C, 256, 256) -> (6, 256, 256, 8) channel-interleaved.
// One block per (face, v) row. Async path: scatter each channel's b32
// straight into interleaved LDS (no VGPR staging), then drain the 8 KB row
// to global as contiguous async b128 stores. Fallback: plain VGPR copy.
// ---------------------------------------------------------------------------
__global__ void cubemap_repack_rows(const float* __restrict__ tex,
                                    float* __restrict__ rp) {
    __shared__ float smem[LRES * PADC];                 // 8 KB interleaved row
    const int row  = blockIdx.x;                        // face*256 + v
    const int u    = threadIdx.x;                       // 0..255
    const int face = row >> 8;
    const int v    = row & (LRES - 1);
    const float* srow = tex + (size_t)face * (CCH * LRES * LRES)
                            + (size_t)v * LRES + u;
    float* drow = rp + (size_t)row * (LRES * PADC);     // 8 KB, 32 B aligned

#if defined(__HIP_DEVICE_COMPILE__) && GFX1250_HAS_ASYNC
    AS3 float* lds = (AS3 float*)smem;
    #pragma unroll
    for (int c = 0; c < CCH; ++c) {
        __builtin_amdgcn_global_load_async_to_lds_b32(
            (AS1 int*)(srow + (size_t)c * (LRES * LRES)),
            (AS3 int*)(lds + u * PADC + c),
            0, 0);
    }
    smem[u * PADC + 6] = 0.0f;                          // pad channels
    smem[u * PADC + 7] = 0.0f;
    __builtin_amdgcn_s_wait_asynccnt(0);
    __syncthreads();
    #pragma unroll
    for (int k = 0; k < 2; ++k) {
        const int chunk = u + k * LRES;                 // 16 B chunk id, 0..511
        __builtin_amdgcn_global_store_async_from_lds_b128(
            (AS1 i32x4*)(drow + chunk * 4),
            (AS3 i32x4*)(lds + chunk * 4),
            0, 0);
    }
    // S_ENDPGM performs an implicit wait-idle, draining ASYNCcnt.
#else
    (void)smem;
    v4f a, b;
    a.x = srow[0 * 65536]; a.y = srow[1 * 65536];
    a.z = srow[2 * 65536]; a.w = srow[3 * 65536];
    b.x = srow[4 * 65536]; b.y = srow[5 * 65536];
    b.z = 0.0f;            b.w = 0.0f;
    v4f* dst = (v4f*)(drow + (size_t)u * PADC);
    dst[0] = a;
    dst[1] = b;
#endif
}

// ---------------------------------------------------------------------------
// Shared direction -> (face, ma, sc, tc) math, matching the reference exactly.
// ---------------------------------------------------------------------------
__device__ __forceinline__ void cube_face(float x, float y, float z,
                                          int& face, float& ma,
                                          float& sc, float& tc) {
    float ax = fabsf(x), ay = fabsf(y), az = fabsf(z);
    bool is_x = (ax >= ay) && (ax >= az);
    bool is_y = !is_x && (ay >= az);
    if (is_x) {
        face = (x >= 0.0f) ? 0 : 1;
        ma = ax;
        sc = (x >= 0.0f) ? -z : z;
        tc = -y;
    } else if (is_y) {
        face = (y >= 0.0f) ? 2 : 3;
        ma = ay;
        sc = x;
        tc = (y >= 0.0f) ? z : -z;
    } else {
        face = (z >= 0.0f) ? 4 : 5;
        ma = az;
        sc = (z >= 0.0f) ? x : -x;
        tc = -y;
    }
}

__device__ __forceinline__ void bilinear_setup(float ma, float sc, float tc,
                                               int& u0, int& u1, int& v0, int& v1,
                                               float& wu, float& wv) {
    float inv = 1.0f / fmaxf(ma, 1e-12f);          // IEEE div: match jnp precision
    float u = 0.5f * (sc * inv + 1.0f);
    float v = 0.5f * (tc * inv + 1.0f);
    float pu = u * (float)LRES - 0.5f;
    float pv = v * (float)LRES - 0.5f;
    float u0f = floorf(pu), v0f = floorf(pv);
    wu = pu - u0f;
    wv = pv - v0f;
    u0 = clampi((int)u0f, 0, LRES - 1);
    u1 = clampi(u0 + 1, 0, LRES - 1);
    v0 = clampi((int)v0f, 0, LRES - 1);
    v1 = clampi(v0 + 1, 0, LRES - 1);
}

// ---------------------------------------------------------------------------
// Pass 2 (fast path): bilinear gather from the repacked, L2-resident texture.
// Streams: directions in (NT loads), output out (NT stores) -> protect the
// texture's L2 residency. Gathers: 2x b128 per corner, 32 B aligned.
// ---------------------------------------------------------------------------
__global__ void cubemap_bilinear(const float* __restrict__ in,
                                 const float* __restrict__ rp,
                                 const float* __restrict__ failv,
                                 float* __restrict__ out, int n) {
    int i = blockIdx.x * blockDim.x + threadIdx.x;
    if (i >= n) return;

    const float* p = in + (size_t)i * 3;
    float x = __builtin_nontemporal_load(p + 0);
    float y = __builtin_nontemporal_load(p + 1);
    float z = __builtin_nontemporal_load(p + 2);

    int face; float ma, sc, tc;
    cube_face(x, y, z, face, ma, sc, tc);

    int u0, u1, v0, v1; float wu, wv;
    bilinear_setup(ma, sc, tc, u0, u1, v0, v1, wu, wv);

    v4f accA, accB;
    if (ma > 0.0f) {
        size_t fbase = (size_t)face << 16;            // face * 256*256 texels
        const v4f* rpv = (const v4f*)rp;
        const v4f* t00 = rpv + ((fbase + (size_t)(v0 * LRES + u0)) << 1);
        const v4f* t10 = rpv + ((fbase + (size_t)(v0 * LRES + u1)) << 1);
        const v4f* t01 = rpv + ((fbase + (size_t)(v1 * LRES + u0)) << 1);
        const v4f* t11 = rpv + ((fbase + (size_t)(v1 * LRES + u1)) << 1);

        v4f a00 = t00[0], b00 = t00[1];
        v4f a10 = t10[0], b10 = t10[1];
        v4f a01 = t01[0], b01 = t01[1];
        v4f a11 = t11[0], b11 = t11[1];

        float w00 = (1.0f - wu) * (1.0f - wv);
        float w10 = wu * (1.0f - wv);
        float w01 = (1.0f - wu) * wv;
        float w11 = wu * wv;

        accA = a00 * w00 + a10 * w10 + a01 * w01 + a11 * w11;
        accB = b00 * w00 + b10 * w10 + b01 * w01 + b11 * w11;
    } else {
        accA.x = failv[0]; accA.y = failv[1]; accA.z = failv[2]; accA.w = failv[3];
        accB.x = failv[4]; accB.y = failv[5]; accB.z = 0.0f;     accB.w = 0.0f;
    }

    float* o = out + (size_t)i * CCH;                 // i*24 B -> 8 B aligned
    v2f s0; s0.x = accA.x; s0.y = accA.y;
    v2f s1; s1.x = accA.z; s1.y = accA.w;
    v2f s2; s2.x = accB.x; s2.y = accB.y;
    __builtin_nontemporal_store(s0, (v2f*)(o + 0));
    __builtin_nontemporal_store(s1, (v2f*)(o + 2));
    __builtin_nontemporal_store(s2, (v2f*)(o + 4));
}

// ---------------------------------------------------------------------------
// Fallback: direct gather from the original (6, C, 256, 256) layout
// (used only if the workspace is too small for the repacked texture).
// ---------------------------------------------------------------------------
__global__ void cubemap_direct(const float* __restrict__ in,
                               const float* __restrict__ tex,
                               const float* __restrict__ failv,
                               float* __restrict__ out, int n) {
    int i = blockIdx.x * blockDim.x + threadIdx.x;
    if (i >= n) return;

    const float* p = in + (size_t)i * 3;
    float x = __builtin_nontemporal_load(p + 0);
    float y = __builtin_nontemporal_load(p + 1);
    float z = __builtin_nontemporal_load(p + 2);

    int face; float ma, sc, tc;
    cube_face(x, y, z, face, ma, sc, tc);

    int u0, u1, v0, v1; float wu, wv;
    bilinear_setup(ma, sc, tc, u0, u1, v0, v1, wu, wv);

    float res[CCH];
    if (ma > 0.0f) {
        float w00 = (1.0f - wu) * (1.0f - wv);
        float w10 = wu * (1.0f - wv);
        float w01 = (1.0f - wu) * wv;
        float w11 = wu * wv;
        size_t fb = (size_t)face * (CCH * LRES * LRES);
        size_t o00 = (size_t)(v0 * LRES + u0);
        size_t o10 = (size_t)(v0 * LRES + u1);
        size_t o01 = (size_t)(v1 * LRES + u0);
        size_t o11 = (size_t)(v1 * LRES + u1);
        #pragma unroll
        for (int c = 0; c < CCH; ++c) {
            const float* pl = tex + fb + (size_t)c * (LRES * LRES);
            res[c] = pl[o00] * w00 + pl[o10] * w10 + pl[o01] * w01 + pl[o11] * w11;
        }
    } else {
        #pragma unroll
        for (int c = 0; c < CCH; ++c) res[c] = failv[c];
    }

    float* o = out + (size_t)i * CCH;
    #pragma unroll
    for (int c = 0; c < CCH; c += 2) {
        v2f s; s.x = res[c]; s.y = res[c + 1];
        __builtin_nontemporal_store(s, (v2f*)(o + c));
    }
}

// ---------------------------------------------------------------------------
// Host entry
// ---------------------------------------------------------------------------
extern "C" void kernel_launch(void* const* d_in, const int* in_sizes, int n_in,
                              void* d_out, int out_size, void* d_ws, size_t ws_size,
                              hipStream_t stream) {
    const float* inputs = (const float*)d_in[0];   // (B, 3) f32
    const float* tex    = (const float*)d_in[1];   // (6, C, 256, 256) f32
    const float* failv  = (const float*)d_in[2];   // (C,) f32
    float*       out    = (float*)d_out;           // (B, C) f32

    const int n = in_sizes[0] / 3;
    const size_t need = (size_t)NPIX * PADC * sizeof(float);   // 12.58 MB

    const int TB = 256;                            // 8 wave32 waves per block
    if (ws_size >= need) {
        float* rp = (float*)d_ws;
        cubemap_repack_rows<<<NFACE * LRES, LRES, 0, stream>>>(tex, rp);
        cubemap_bilinear<<<(n + TB - 1) / TB, TB, 0, stream>>>(inputs, rp, failv, out, n);
    } else {
        cubemap_direct<<<(n + TB - 1) / TB, TB, 0, stream>>>(inputs, tex, failv, out, n);
    }
}